// SharedTNNActorCritic_88381837017464
// MI455X (gfx1250) — compile-verified
//
#include <hip/hip_runtime.h>
#include <math.h>

// ---------------- problem constants (from reference) ----------------
#define N0v 3072
#define N1v 6144
#define N2v 3072
#define RRv 64
#define EEv 32
#define HHv 128
#define DDv 200

// Flattened params offsets (JAX pytree: dict keys sorted alphabetically)
#define OFF_AFC1_B 0
#define OFF_AFC1_W 128
#define OFF_AFC2_B 4224
#define OFF_AFC2_W 4352
#define OFF_CFC1_B 20736
#define OFF_CFC1_W 20864
#define OFF_CFC2_B 24960
#define OFF_CFC2_W 25088
#define OFF_DIR_B  41472
#define OFF_DIR_W  41672
#define OFF_LAYERS 67272
#define OFF_LIN0_B 67416
#define OFF_LIN0_W 67448
#define OFF_MAG_B  67544
#define OFF_MAG_W  67545
#define OFF_VAL_B  67673
#define OFF_VAL_W  67674

typedef __attribute__((ext_vector_type(2))) float v2f;
typedef __attribute__((ext_vector_type(8))) float v8f;

#define KCHUNK_MAX 384            // K/16 for K in {3072,6144}
#define SB_STRIDE  (KCHUNK_MAX + 2)

// =====================================================================
// Core bandwidth kernel: Y[M,3] += A[M,K] @ T[K,3]   (f32 WMMA 16x16x4)
// Block = 256 threads = 8 waves; each wave owns one 16-row M tile.
// grid = (M/128, KSPLIT); kchunk = K/KSPLIT (divisible by 16, <= 384).
// T chunk staged zero-padded+transposed in LDS -> branch-free b64 B-frag
// loads. A streamed straight from HBM with a 4-deep pipelined fragment
// preload + global_prefetch ahead of the consume point.
// =====================================================================
__global__ void __launch_bounds__(256)
wmma_gemm3_acc(const float* __restrict__ A, const float* __restrict__ T,
               float* __restrict__ Y, int M, int K, int kchunk)
{
    // sB[n][k]: B transposed, 16 rows (rows 3..15 are zeros), padded stride
    __shared__ float sB[16 * SB_STRIDE];
    const int tid  = threadIdx.x;
    const int wave = tid >> 5;
    const int lane = tid & 31;
    const int kbase = blockIdx.y * kchunk;

    // stage: sB[n][k] = (n<3) ? T[kbase+k][n] : 0   (branch-free WMMA loop after)
#pragma unroll
    for (int nn = 0; nn < 16; ++nn) {
        if (nn < 3) {
            for (int k = tid; k < kchunk; k += 256)
                sB[nn * SB_STRIDE + k] = T[(size_t)(kbase + k) * 3 + nn];
        } else {
            for (int k = tid; k < kchunk; k += 256)
                sB[nn * SB_STRIDE + k] = 0.f;
        }
    }
    __syncthreads();

    const int m0    = blockIdx.x * 128 + wave * 16;
    const int n     = lane & 15;               // output column (only n<3 useful)
    const int khalf = (lane >> 4) << 1;        // lanes 0-15: K=k,k+1 ; 16-31: K=k+2,k+3
    const float* Ar  = A + (size_t)(m0 + n) * (size_t)K + kbase + khalf;
    const float* sRow = &sB[n * SB_STRIDE + khalf];

    v8f acc = {0.f, 0.f, 0.f, 0.f, 0.f, 0.f, 0.f, 0.f};

    // 4-deep pipelined fragment preload (group of 16 K per outer iteration)
    v2f a0 = *(const v2f*)(Ar + 0);   v2f b0 = *(const v2f*)(sRow + 0);
    v2f a1 = *(const v2f*)(Ar + 4);   v2f b1 = *(const v2f*)(sRow + 4);
    v2f a2 = *(const v2f*)(Ar + 8);   v2f b2 = *(const v2f*)(sRow + 8);
    v2f a3 = *(const v2f*)(Ar + 12);  v2f b3 = *(const v2f*)(sRow + 12);

    for (int kk = 16; kk < kchunk; kk += 16) {
        __builtin_prefetch(Ar + kk + 128, 0, 1);   // global_prefetch ~512B ahead
        v2f na0 = *(const v2f*)(Ar + kk + 0);   v2f nb0 = *(const v2f*)(sRow + kk + 0);
        v2f na1 = *(const v2f*)(Ar + kk + 4);   v2f nb1 = *(const v2f*)(sRow + kk + 4);
        v2f na2 = *(const v2f*)(Ar + kk + 8);   v2f nb2 = *(const v2f*)(sRow + kk + 8);
        v2f na3 = *(const v2f*)(Ar + kk + 12);  v2f nb3 = *(const v2f*)(sRow + kk + 12);
        acc = __builtin_amdgcn_wmma_f32_16x16x4_f32(false, a0, false, b0, (short)0, acc, false, false);
        acc = __builtin_amdgcn_wmma_f32_16x16x4_f32(false, a1, false, b1, (short)0, acc, false, false);
        acc = __builtin_amdgcn_wmma_f32_16x16x4_f32(false, a2, false, b2, (short)0, acc, false, false);
        acc = __builtin_amdgcn_wmma_f32_16x16x4_f32(false, a3, false, b3, (short)0, acc, false, false);
        a0 = na0; b0 = nb0; a1 = na1; b1 = nb1;
        a2 = na2; b2 = nb2; a3 = na3; b3 = nb3;
    }
    acc = __builtin_amdgcn_wmma_f32_16x16x4_f32(false, a0, false, b0, (short)0, acc, false, false);
    acc = __builtin_amdgcn_wmma_f32_16x16x4_f32(false, a1, false, b1, (short)0, acc, false, false);
    acc = __builtin_amdgcn_wmma_f32_16x16x4_f32(false, a2, false, b2, (short)0, acc, false, false);
    acc = __builtin_amdgcn_wmma_f32_16x16x4_f32(false, a3, false, b3, (short)0, acc, false, false);

    // D layout: VGPR v, lanes 0-15 -> M=v ; lanes 16-31 -> M=v+8 ; N=lane%16
    if (n < 3) {
        const int rbase = m0 + ((lane >> 4) << 3);
#pragma unroll
        for (int v = 0; v < 8; ++v)
            atomicAdd(&Y[(size_t)(rbase + v) * 3 + n], acc[v]);
    }
}

// ---------------- tiny helper kernels ----------------

// t[i,:] = x[i,:] @ W  (W is 3x3 row-major [in,out])
__global__ void xform3(const float* __restrict__ x, const float* __restrict__ W,
                       float* __restrict__ t, int N)
{
    int i = blockIdx.x * blockDim.x + threadIdx.x;
    if (i >= N) return;
    float a0 = x[3 * i], a1 = x[3 * i + 1], a2 = x[3 * i + 2];
#pragma unroll
    for (int j = 0; j < 3; ++j)
        t[3 * i + j] = a0 * W[j] + a1 * W[3 + j] + a2 * W[6 + j];
}

__global__ void sigmoid_k(const float* __restrict__ y, float* __restrict__ x, int n)
{
    int i = blockIdx.x * blockDim.x + threadIdx.x;
    if (i < n) x[i] = 1.f / (1.f + __expf(-y[i]));
}

// masked mean over nodes: reg[r,e]
__global__ void pool_k(const float* __restrict__ x0e,
                       const unsigned char* __restrict__ mask,
                       float* __restrict__ reg)
{
    int idx = blockIdx.x * blockDim.x + threadIdx.x;
    if (idx >= RRv * EEv) return;
    int r = idx / EEv, e = idx % EEv;
    const unsigned char* mrow = mask + (size_t)r * N0v;
    float s = 0.f; int cnt = 0;
    for (int i = 0; i < N0v; ++i) {
        if (mrow[i]) { s += x0e[(size_t)i * EEv + e]; ++cnt; }
    }
    reg[idx] = (cnt > 0) ? s / (float)cnt : 0.f;
}

// generic FC: out[r, ooff+o (stride ostride)] = act(b[o] + in[r,:] @ w[:,o])
// act: 0=none 1=relu 2=tanh
__global__ void fc_k(const float* __restrict__ in, const float* __restrict__ w,
                     const float* __restrict__ b, float* __restrict__ out,
                     int R_, int fin, int fout, int ostride, int ooff, int act)
{
    int idx = blockIdx.x * blockDim.x + threadIdx.x;
    if (idx >= R_ * fout) return;
    int r = idx / fout, o = idx % fout;
    float s = b[o];
    const float* ir = in + (size_t)r * fin;
    for (int i = 0; i < fin; ++i) s += ir[i] * w[(size_t)i * fout + o];
    if (act == 1) s = fmaxf(s, 0.f);
    else if (act == 2) s = tanhf(s);
    out[(size_t)r * ostride + ooff + o] = s;
}

// =====================================================================
extern "C" void kernel_launch(void* const* d_in, const int* in_sizes, int n_in,
                              void* d_out, int out_size, void* d_ws, size_t ws_size,
                              hipStream_t stream)
{
    const float* x0    = (const float*)d_in[0];
    const float* x1    = (const float*)d_in[1];
    const float* x2    = (const float*)d_in[2];
    const float* inc1  = (const float*)d_in[3];   // [N0,N1]
    const float* inc1n = (const float*)d_in[4];   // [N1,N0]
    const float* inc2  = (const float*)d_in[5];   // [N1,N2]
    const float* inc2n = (const float*)d_in[6];   // [N2,N1]
    const float* au0   = (const float*)d_in[7];   // [N0,N0]
    const float* au1   = (const float*)d_in[8];   // [N1,N1]
    const float* ad1   = (const float*)d_in[9];   // [N1,N1]
    const float* ad2   = (const float*)d_in[10];  // [N2,N2]
    const unsigned char* masks = (const unsigned char*)d_in[11]; // [R,N0] bool
    const float* params = (const float*)d_in[12];

    // ---- workspace carve-up (all fully written every call) ----
    float* p = (float*)d_ws;
    auto alloc = [&](size_t nf) { float* r = p; p += nf; return r; };
    float* t00  = alloc((size_t)N0v * 3); float* t01  = alloc((size_t)N0v * 3);
    float* t10  = alloc((size_t)N1v * 3); float* t11u = alloc((size_t)N1v * 3);
    float* t11d = alloc((size_t)N1v * 3); float* t12  = alloc((size_t)N1v * 3);
    float* t21  = alloc((size_t)N2v * 3); float* t22  = alloc((size_t)N2v * 3);
    float* y0   = alloc((size_t)N0v * 3);
    float* y1   = alloc((size_t)N1v * 3);
    float* y2   = alloc((size_t)N2v * 3);
    float* xa0  = alloc((size_t)N0v * 3);
    float* xa1  = alloc((size_t)N1v * 3);
    float* xa2  = alloc((size_t)N2v * 3);
    float* x0e  = alloc((size_t)N0v * EEv);
    float* reg  = alloc((size_t)RRv * EEv);
    float* h1   = alloc((size_t)RRv * HHv);
    float* h2   = alloc((size_t)RRv * HHv);
    float* c1   = alloc((size_t)RRv * HHv);
    float* c2   = alloc((size_t)RRv * HHv);

    const int KS = 16;  // K-split; kchunk = K/16 in {192,384}
    const int tb = 256;

    const float* cur0 = x0; const float* cur1 = x1; const float* cur2 = x2;
    for (int l = 0; l < 2; ++l) {
        const float* LW = params + OFF_LAYERS + l * 72;
        const float *W00 = LW, *W01 = LW + 9, *W10 = LW + 18, *W11d = LW + 27,
                    *W11u = LW + 36, *W12 = LW + 45, *W21 = LW + 54, *W22 = LW + 63;

        xform3<<<(N0v + tb - 1) / tb, tb, 0, stream>>>(cur0, W00,  t00,  N0v);
        xform3<<<(N0v + tb - 1) / tb, tb, 0, stream>>>(cur0, W01,  t01,  N0v);
        xform3<<<(N1v + tb - 1) / tb, tb, 0, stream>>>(cur1, W10,  t10,  N1v);
        xform3<<<(N1v + tb - 1) / tb, tb, 0, stream>>>(cur1, W11u, t11u, N1v);
        xform3<<<(N1v + tb - 1) / tb, tb, 0, stream>>>(cur1, W11d, t11d, N1v);
        xform3<<<(N1v + tb - 1) / tb, tb, 0, stream>>>(cur1, W12,  t12,  N1v);
        xform3<<<(N2v + tb - 1) / tb, tb, 0, stream>>>(cur2, W21,  t21,  N2v);
        xform3<<<(N2v + tb - 1) / tb, tb, 0, stream>>>(cur2, W22,  t22,  N2v);

        hipMemsetAsync(y0, 0, (size_t)N0v * 3 * sizeof(float), stream);
        hipMemsetAsync(y1, 0, (size_t)N1v * 3 * sizeof(float), stream);
        hipMemsetAsync(y2, 0, (size_t)N2v * 3 * sizeof(float), stream);

        // y0 = au0@t00 + inc1@t10
        wmma_gemm3_acc<<<dim3(N0v / 128, KS), 256, 0, stream>>>(au0,   t00,  y0, N0v, N0v, N0v / KS);
        wmma_gemm3_acc<<<dim3(N0v / 128, KS), 256, 0, stream>>>(inc1,  t10,  y0, N0v, N1v, N1v / KS);
        // y1 = inc1n@t01 + au1@t11u + ad1@t11d + inc2@t21
        wmma_gemm3_acc<<<dim3(N1v / 128, KS), 256, 0, stream>>>(inc1n, t01,  y1, N1v, N0v, N0v / KS);
        wmma_gemm3_acc<<<dim3(N1v / 128, KS), 256, 0, stream>>>(au1,   t11u, y1, N1v, N1v, N1v / KS);
        wmma_gemm3_acc<<<dim3(N1v / 128, KS), 256, 0, stream>>>(ad1,   t11d, y1, N1v, N1v, N1v / KS);
        wmma_gemm3_acc<<<dim3(N1v / 128, KS), 256, 0, stream>>>(inc2,  t21,  y1, N1v, N2v, N2v / KS);
        // y2 = inc2n@t12 + ad2@t22
        wmma_gemm3_acc<<<dim3(N2v / 128, KS), 256, 0, stream>>>(inc2n, t12,  y2, N2v, N1v, N1v / KS);
        wmma_gemm3_acc<<<dim3(N2v / 128, KS), 256, 0, stream>>>(ad2,   t22,  y2, N2v, N2v, N2v / KS);

        sigmoid_k<<<(N0v * 3 + 255) / 256, 256, 0, stream>>>(y0, xa0, N0v * 3);
        sigmoid_k<<<(N1v * 3 + 255) / 256, 256, 0, stream>>>(y1, xa1, N1v * 3);
        sigmoid_k<<<(N2v * 3 + 255) / 256, 256, 0, stream>>>(y2, xa2, N2v * 3);
        cur0 = xa0; cur1 = xa1; cur2 = xa2;
    }

    // node embeddings: x0e = x0 @ lin0_w + lin0_b   [N0, E]
    fc_k<<<(N0v * EEv + 255) / 256, 256, 0, stream>>>(
        cur0, params + OFF_LIN0_W, params + OFF_LIN0_B, x0e, N0v, 3, EEv, EEv, 0, 0);

    // region pooling
    pool_k<<<(RRv * EEv + 255) / 256, 256, 0, stream>>>(x0e, masks, reg);

    float* out = (float*)d_out;
    // actor head
    fc_k<<<(RRv * HHv + 255) / 256, 256, 0, stream>>>(
        reg, params + OFF_AFC1_W, params + OFF_AFC1_B, h1, RRv, EEv, HHv, HHv, 0, 1);
    fc_k<<<(RRv * HHv + 255) / 256, 256, 0, stream>>>(
        h1, params + OFF_AFC2_W, params + OFF_AFC2_B, h2, RRv, HHv, HHv, HHv, 0, 1);
    fc_k<<<(RRv * DDv + 255) / 256, 256, 0, stream>>>(
        h2, params + OFF_DIR_W, params + OFF_DIR_B, out, RRv, HHv, DDv, DDv + 2, 0, 0);
    fc_k<<<(RRv + 255) / 256, 256, 0, stream>>>(
        h2, params + OFF_MAG_W, params + OFF_MAG_B, out, RRv, HHv, 1, DDv + 2, DDv, 2);
    // critic head
    fc_k<<<(RRv * HHv + 255) / 256, 256, 0, stream>>>(
        reg, params + OFF_CFC1_W, params + OFF_CFC1_B, c1, RRv, EEv, HHv, HHv, 0, 1);
    fc_k<<<(RRv * HHv + 255) / 256, 256, 0, stream>>>(
        c1, params + OFF_CFC2_W, params + OFF_CFC2_B, c2, RRv, HHv, HHv, HHv, 0, 1);
    fc_k<<<(RRv + 255) / 256, 256, 0, stream>>>(
        c2, params + OFF_VAL_W, params + OFF_VAL_B, out, RRv, HHv, 1, DDv + 2, DDv + 1, 0);
}